// DCWTv2Attention_57629871178541
// MI455X (gfx1250) — compile-verified
//
#include <hip/hip_runtime.h>
#include <math.h>

// ---- problem constants (match reference) ----
#define B_    2
#define N_    1024
#define E_    512
#define H_    8
#define DH_   64
#define KMAX_ 8
#define WIN_  32
#define DP_   12
#define NODES_ 2047   // sum_{d=0..10} N>>d

typedef float v2f __attribute__((ext_vector_type(2)));
typedef float v8f __attribute__((ext_vector_type(8)));

__device__ __forceinline__ float wsum(float v) {
#pragma unroll
  for (int o = 16; o > 0; o >>= 1) v += __shfl_xor(v, o, 32);
  return v;
}
__device__ __forceinline__ float wmaxr(float v) {
#pragma unroll
  for (int o = 16; o > 0; o >>= 1) v = fmaxf(v, __shfl_xor(v, o, 32));
  return v;
}
__device__ __forceinline__ float sigmoidf_(float x) { return 1.f / (1.f + __expf(-x)); }

// =====================================================================
// WMMA fp32 GEMM:  Y[m, n] = sum_k X[m, k] * W[n, k] + bias[n]
// One wave computes one 16x16 C tile via V_WMMA_F32_16X16X4_F32.
// A frag (16x4, ISA 7.12.2): lane m = lane&15, k = 2*(lane>>4)+{0,1}
// B frag (4x16):             lane n = lane&15, k = 2*(lane>>4)+{0,1}
// C frag: VGPR i -> M = i + 8*(lane>>4), N = lane&15
// =====================================================================
__global__ void proj_gemm(const float* __restrict__ X, const float* __restrict__ W,
                          const float* __restrict__ bias, float* __restrict__ Y,
                          int Kdim, int NC) {
  const int wave = threadIdx.x >> 5, lane = threadIdx.x & 31;
  const int r = lane & 15, hl = lane >> 4;
  const int m0 = blockIdx.x * 16;
  const int n0 = blockIdx.y * 64 + wave * 16;
  v8f acc = {0.f, 0.f, 0.f, 0.f, 0.f, 0.f, 0.f, 0.f};
  const float* xr = X + (size_t)(m0 + r) * Kdim + 2 * hl;
  const float* wr = W + (size_t)(n0 + r) * Kdim + 2 * hl;
  for (int k0 = 0; k0 < Kdim; k0 += 4) {
    v2f a = *(const v2f*)(xr + k0);
    v2f b = *(const v2f*)(wr + k0);
    acc = __builtin_amdgcn_wmma_f32_16x16x4_f32(false, a, false, b, (short)0, acc,
                                                false, false);
  }
  const int col = n0 + r;
  const float bb = bias[col];
#pragma unroll
  for (int i = 0; i < 8; ++i) {
    const int row = m0 + i + 8 * hl;
    Y[(size_t)row * NC + col] = acc[i] + bb;
  }
}

// Qall[p] = q @ ddqW[p]^T + q   (q viewed as (B*N*H, DH))
__global__ void qall_gemm(const float* __restrict__ q, const float* __restrict__ ddqW,
                          float* __restrict__ Qall) {
  const int wave = threadIdx.x >> 5, lane = threadIdx.x & 31;
  const int r = lane & 15, hl = lane >> 4;
  const int m0 = blockIdx.x * 16;
  const int p  = blockIdx.y;
  const int n0 = wave * 16;
  v8f acc = {0.f, 0.f, 0.f, 0.f, 0.f, 0.f, 0.f, 0.f};
  const float* xr = q + (size_t)(m0 + r) * DH_ + 2 * hl;
  const float* wr = ddqW + ((size_t)p * DH_ + (n0 + r)) * DH_ + 2 * hl;
#pragma unroll
  for (int k0 = 0; k0 < DH_; k0 += 4) {
    v2f a = *(const v2f*)(xr + k0);
    v2f b = *(const v2f*)(wr + k0);
    acc = __builtin_amdgcn_wmma_f32_16x16x4_f32(false, a, false, b, (short)0, acc,
                                                false, false);
  }
  const int col = n0 + r;
  float* out = Qall + (size_t)p * ((size_t)B_ * N_ * H_ * DH_);
#pragma unroll
  for (int i = 0; i < 8; ++i) {
    const int row = m0 + i + 8 * hl;
    out[(size_t)row * DH_ + col] = acc[i] + q[(size_t)row * DH_ + col];
  }
}

// =====================================================================
// Level 0 of the tree bank: slot 0 = v, slots 1..7 = 0
// bank layout: bank[(((b*H+h)*NODES + idx)*KMAX + k)*DH + dd]
// =====================================================================
__global__ void scatter_leaves(const float* __restrict__ v, float* __restrict__ bank) {
  const int t = blockIdx.x * blockDim.x + threadIdx.x; // over B*N*E
  const int e = t % E_;
  const int n = (t / E_) % N_;
  const int b = t / (E_ * N_);
  const int h = e >> 6, dd = e & 63;
  const size_t base = (((size_t)(b * H_ + h) * NODES_ + n) * KMAX_) * DH_ + dd;
  bank[base] = v[t];
#pragma unroll
  for (int k = 1; k < KMAX_; ++k) bank[base + (size_t)k * DH_] = 0.f;
}

// =====================================================================
// Windowed causal local attention: one wave per (b,n,h), lane = window pos
// =====================================================================
__global__ void local_attn(const float* __restrict__ q, const float* __restrict__ kl,
                           const float* __restrict__ vl, float* __restrict__ local) {
  const int n = blockIdx.x, b = blockIdx.y;
  const int h = threadIdx.x >> 5, lane = threadIdx.x & 31;
  const int j = n - lane;
  const float* qp = q + ((size_t)(b * N_ + n)) * E_ + h * DH_;
  float s;
  if (j >= 0) {
    const float* kp = kl + ((size_t)(b * N_ + j)) * E_ + h * DH_;
    s = 0.f;
#pragma unroll 8
    for (int d2 = 0; d2 < DH_; ++d2) s += qp[d2] * kp[d2];
    s *= 0.125f; // 1/sqrt(64)
  } else {
    s = -1e30f;
  }
  const float mx = wmaxr(s);
  const float ev = (j >= 0) ? __expf(s - mx) : 0.f;
  const float sm = wsum(ev);
  const float pw_self = ev / sm;
  const int dd0 = 2 * lane;
  float ox = 0.f, oy = 0.f;
#pragma unroll
  for (int w = 0; w < WIN_; ++w) {
    const float pw = __shfl(pw_self, w, 32);
    const int jj = n - w;
    if (jj >= 0) {
      const float* vp = vl + ((size_t)(b * N_ + jj)) * E_ + h * DH_ + dd0;
      ox += pw * vp[0];
      oy += pw * vp[1];
    }
  }
  float* op = local + ((size_t)(b * N_ + n)) * E_ + h * DH_ + dd0;
  op[0] = ox;
  op[1] = oy;
}

// =====================================================================
// One tree level d: rotate/gate/attention-pool/LN/skip/head-couple
// Block = 256 threads = 8 waves = 8 heads; grid (nn, B).
// Templated on KPREV (k of level d-1) and KP (k of level d) so all
// register arrays have static indices.
// =====================================================================
template <int KPREV, int KP>
__global__ void level_kernel(float* __restrict__ bank, int d, int off_prev, int off_cur,
                             const float* __restrict__ wfreq, const float* __restrict__ wdamp,
                             const float* __restrict__ wphase,
                             const float* __restrict__ glW, const float* __restrict__ glb,
                             const float* __restrict__ grW, const float* __restrict__ grb,
                             const float* __restrict__ pq, const float* __restrict__ lnG,
                             const float* __restrict__ lnB, const float* __restrict__ skA,
                             const float* __restrict__ skW, const float* __restrict__ coup) {
  constexpr int K2N = 2 * KPREV;
  __shared__ float gin_sh[H_][2 * DH_];
  __shared__ float bank_sh[H_][16][DH_];
  __shared__ float scr_sh[H_][KMAX_][16];
  __shared__ float att_sh[H_][KMAX_][16];
  __shared__ float par_sh[H_][KMAX_][DH_];

  const int n = blockIdx.x, b = blockIdx.y;
  const int h = threadIdx.x >> 5, lane = threadIdx.x & 31;
  const int dd0 = 2 * lane, dd1 = dd0 + 1;
  const bool lo = lane < 16;

  // per-head rotation constants: dec = exp(-softplus(wdamp)); ang = wf + wp + d*pi/4
  const float alpha = log1pf(__expf(wdamp[h]));
  const float dec = __expf(-alpha);
  const float ang = wfreq[h] + wphase[h] + (float)d * 0.7853981633974483f;
  const float pr = dec * __cosf(ang);
  const float pim = dec * __sinf(ang);

  const size_t baseL = (((size_t)(b * H_ + h) * NODES_ + off_prev + 2 * n) * KMAX_) * DH_;
  const size_t baseR = baseL + (size_t)KMAX_ * DH_;

  float fLx[KPREV], fLy[KPREV], rox[KPREV], roy[KPREV];
#pragma unroll
  for (int k = 0; k < KPREV; ++k) {
    fLx[k] = bank[baseL + k * DH_ + dd0];
    fLy[k] = bank[baseL + k * DH_ + dd1];
    const float rx = bank[baseR + k * DH_ + dd0];
    const float ry = bank[baseR + k * DH_ + dd1];
    const float px = __shfl_xor(rx, 16, 32); // partner at dd ^ 32
    const float py = __shfl_xor(ry, 16, 32);
    rox[k] = lo ? (pr * rx - pim * px) : (pim * px + pr * rx);
    roy[k] = lo ? (pr * ry - pim * py) : (pim * py + pr * ry);
  }

  // means over k
  float lmx = 0.f, lmy = 0.f, rmx = 0.f, rmy = 0.f;
#pragma unroll
  for (int k = 0; k < KPREV; ++k) { lmx += fLx[k]; lmy += fLy[k]; rmx += rox[k]; rmy += roy[k]; }
  const float invk = 1.f / (float)KPREV;
  lmx *= invk; lmy *= invk; rmx *= invk; rmy *= invk;
  gin_sh[h][dd0] = lmx;        gin_sh[h][dd1] = lmy;
  gin_sh[h][DH_ + dd0] = rmx;  gin_sh[h][DH_ + dd1] = rmy;
  __syncthreads();

  // gates gl, gr : 64x128 matvec each (this lane's two output channels)
  float aLx = glb[d * DH_ + dd0], aLy = glb[d * DH_ + dd1];
  float aRx = grb[d * DH_ + dd0], aRy = grb[d * DH_ + dd1];
  const float* glw0 = glW + (size_t)(d * DH_ + dd0) * (2 * DH_);
  const float* glw1 = glW + (size_t)(d * DH_ + dd1) * (2 * DH_);
  const float* grw0 = grW + (size_t)(d * DH_ + dd0) * (2 * DH_);
  const float* grw1 = grW + (size_t)(d * DH_ + dd1) * (2 * DH_);
#pragma unroll 4
  for (int f = 0; f < 2 * DH_; ++f) {
    const float g = gin_sh[h][f];
    aLx += g * glw0[f]; aLy += g * glw1[f];
    aRx += g * grw0[f]; aRy += g * grw1[f];
  }
  const float glx = sigmoidf_(aLx), gly = sigmoidf_(aLy);
  const float grx = sigmoidf_(aRx), gry = sigmoidf_(aRy);

  // gated bank (2*KPREV entries)
  float bkx[K2N], bky[K2N];
#pragma unroll
  for (int k = 0; k < KPREV; ++k) {
    bkx[k] = fLx[k] * glx;          bky[k] = fLy[k] * gly;
    bkx[KPREV + k] = rox[k] * grx;  bky[KPREV + k] = roy[k] * gry;
  }
#pragma unroll
  for (int k = 0; k < K2N; ++k) { bank_sh[h][k][dd0] = bkx[k]; bank_sh[h][k][dd1] = bky[k]; }
  __syncthreads();

  // attention-pool scores: s[qi][k] = dot(pq[d,qi], bank[k]) / 8
  constexpr int TOTAL = KP * K2N;
  for (int t = lane; t < TOTAL; t += 32) {
    const int qi = t / K2N, k = t % K2N;
    const float* pqp = pq + (size_t)(d * KMAX_ + qi) * DH_;
    float s = 0.f;
#pragma unroll 8
    for (int f = 0; f < DH_; ++f) s += pqp[f] * bank_sh[h][k][f];
    scr_sh[h][qi][k] = s * 0.125f;
  }
  __syncthreads();

  if (lane < KP) {
    float mx = -1e30f;
#pragma unroll
    for (int k = 0; k < K2N; ++k) mx = fmaxf(mx, scr_sh[h][lane][k]);
    float sm = 0.f;
    float ev[K2N];
#pragma unroll
    for (int k = 0; k < K2N; ++k) { ev[k] = __expf(scr_sh[h][lane][k] - mx); sm += ev[k]; }
    const float inv = 1.f / sm;
#pragma unroll
    for (int k = 0; k < K2N; ++k) att_sh[h][lane][k] = ev[k] * inv;
  }
  __syncthreads();

  // pooled vectors + LayerNorm + skip
  float prx[KP], pry[KP];
#pragma unroll
  for (int qi = 0; qi < KP; ++qi) {
    float ax = 0.f, ay = 0.f;
#pragma unroll
    for (int k = 0; k < K2N; ++k) {
      const float w = att_sh[h][qi][k];
      ax += w * bkx[k];
      ay += w * bky[k];
    }
    prx[qi] = ax; pry[qi] = ay;
  }
  const float g0 = lnG[d * DH_ + dd0], g1 = lnG[d * DH_ + dd1];
  const float b0 = lnB[d * DH_ + dd0], b1 = lnB[d * DH_ + dd1];
#pragma unroll
  for (int qi = 0; qi < KP; ++qi) {
    const float mu = wsum(prx[qi] + pry[qi]) * (1.f / DH_);
    const float dx = prx[qi] - mu, dy = pry[qi] - mu;
    const float var = wsum(dx * dx + dy * dy) * (1.f / DH_);
    const float inv = rsqrtf(var + 1e-5f);
    prx[qi] = dx * inv * g0 + b0;
    pry[qi] = dy * inv * g1 + b1;
  }
  // skip: sigmoid(skA[d]) * (skW[d] @ lm)
  const float sA = sigmoidf_(skA[d]);
  const float* skw0 = skW + (size_t)(d * DH_ + dd0) * DH_;
  const float* skw1 = skW + (size_t)(d * DH_ + dd1) * DH_;
  float skx = 0.f, sky = 0.f;
#pragma unroll 8
  for (int f = 0; f < DH_; ++f) {
    const float lmv = gin_sh[h][f]; // lm lives in gin[0..63]
    skx += lmv * skw0[f];
    sky += lmv * skw1[f];
  }
#pragma unroll
  for (int qi = 0; qi < KP; ++qi) { prx[qi] += sA * skx; pry[qi] += sA * sky; }

#pragma unroll
  for (int qi = 0; qi < KP; ++qi) { par_sh[h][qi][dd0] = prx[qi]; par_sh[h][qi][dd1] = pry[qi]; }
  __syncthreads();

  // head coupling: cur[i] = sum_j softmax(coup[d])[i,j] * par[j]
  float cw[H_];
  {
    float mx = -1e30f;
#pragma unroll
    for (int jj = 0; jj < H_; ++jj) { cw[jj] = coup[(size_t)(d * H_ + h) * H_ + jj]; mx = fmaxf(mx, cw[jj]); }
    float sm = 0.f;
#pragma unroll
    for (int jj = 0; jj < H_; ++jj) { cw[jj] = __expf(cw[jj] - mx); sm += cw[jj]; }
    const float inv = 1.f / sm;
#pragma unroll
    for (int jj = 0; jj < H_; ++jj) cw[jj] *= inv;
  }
  const size_t baseO = (((size_t)(b * H_ + h) * NODES_ + off_cur + n) * KMAX_) * DH_;
#pragma unroll
  for (int qi = 0; qi < KMAX_; ++qi) {
    float ox = 0.f, oy = 0.f;
    if (qi < KP) {
#pragma unroll
      for (int jj = 0; jj < H_; ++jj) {
        ox += cw[jj] * par_sh[jj][qi][dd0];
        oy += cw[jj] * par_sh[jj][qi][dd1];
      }
    }
    bank[baseO + qi * DH_ + dd0] = ox;
    bank[baseO + qi * DH_ + dd1] = oy;
  }
}

// =====================================================================
// Tree gather attention + gate + combine:  hidden = local + gate*tree
// Block = 8 waves (heads); grid (N, B).
// =====================================================================
__global__ void tree_combine(const float* __restrict__ Qall, const float* __restrict__ bank,
                             const float* __restrict__ local, const float* __restrict__ gpre,
                             const float* __restrict__ ddqT, const int* __restrict__ cov_idx,
                             const int* __restrict__ cov_depth,
                             const unsigned char* __restrict__ cov_mask,
                             const int* __restrict__ kvalid, float* __restrict__ hidden, int S) {
  __shared__ float sc_sh[H_][16 * KMAX_]; // S*KMAX <= 128
  const int n = blockIdx.x, b = blockIdx.y;
  const int h = threadIdx.x >> 5, lane = threadIdx.x & 31;
  const int dd0 = 2 * lane, dd1 = dd0 + 1;
  const int SK = S * KMAX_;

  // pass 1: scores
  for (int s = 0; s < S; ++s) {
    const bool valid = cov_mask[n * S + s] != 0;
    const int p = cov_depth[n * S + s];
    const int idx = cov_idx[n * S + s];
    const int kv = valid ? kvalid[idx] : 0;
    const float scd = 1.f / ((log1pf(__expf(ddqT[p])) + 1e-6f) * 8.f);
    float qx = 0.f, qy = 0.f;
    if (valid) {
      const float* qp = Qall + ((size_t)p * ((size_t)B_ * N_ * H_) +
                                (size_t)(b * N_ + n) * H_ + h) * DH_;
      qx = qp[dd0]; qy = qp[dd1];
    }
#pragma unroll
    for (int k = 0; k < KMAX_; ++k) {
      float sc = -1e30f;
      if (valid && k < kv) {
        const float* bg = bank + (((size_t)(b * H_ + h) * NODES_ + idx) * KMAX_ + k) * DH_;
        float part = qx * bg[dd0] + qy * bg[dd1];
        part = wsum(part);
        sc = part * scd;
      }
      if (lane == 0) sc_sh[h][s * KMAX_ + k] = sc;
    }
  }
  __syncthreads();

  float mx = -1e30f;
  for (int t = 0; t < SK; ++t) mx = fmaxf(mx, sc_sh[h][t]);
  float tx = 0.f, ty = 0.f;
  if (mx > -1e29f) {
    float sm = 0.f;
    for (int t = 0; t < SK; ++t) {
      const float v = sc_sh[h][t];
      if (v > -1e29f) sm += __expf(v - mx);
    }
    const float inv = 1.f / sm;
    for (int s = 0; s < S; ++s) {
      if (!cov_mask[n * S + s]) continue;
      const int idx = cov_idx[n * S + s];
      const int kv = kvalid[idx];
#pragma unroll
      for (int k = 0; k < KMAX_; ++k) {
        if (k < kv) {
          const float w = __expf(sc_sh[h][s * KMAX_ + k] - mx) * inv;
          const float* bg = bank + (((size_t)(b * H_ + h) * NODES_ + idx) * KMAX_ + k) * DH_;
          tx += w * bg[dd0];
          ty += w * bg[dd1];
        }
      }
    }
  }
  const size_t eoff = (size_t)(b * N_ + n) * E_ + h * DH_;
  const float ga = sigmoidf_(gpre[eoff + dd0]);
  const float gb2 = sigmoidf_(gpre[eoff + dd1]);
  hidden[eoff + dd0] = local[eoff + dd0] + ga * tx;
  hidden[eoff + dd1] = local[eoff + dd1] + gb2 * ty;
}

// =====================================================================
// Host launcher
// =====================================================================
extern "C" void kernel_launch(void* const* d_in, const int* in_sizes, int n_in,
                              void* d_out, int out_size, void* d_ws, size_t ws_size,
                              hipStream_t stream) {
  (void)n_in; (void)out_size; (void)ws_size;
  const float* x    = (const float*)d_in[0];
  const float* qW   = (const float*)d_in[1];  const float* qb  = (const float*)d_in[2];
  const float* vW   = (const float*)d_in[3];  const float* vb  = (const float*)d_in[4];
  const float* oW   = (const float*)d_in[5];  const float* ob  = (const float*)d_in[6];
  const float* klW  = (const float*)d_in[7];  const float* klb = (const float*)d_in[8];
  const float* vlW  = (const float*)d_in[9];  const float* vlb = (const float*)d_in[10];
  const float* gW   = (const float*)d_in[11]; const float* gb  = (const float*)d_in[12];
  const float* ddqW = (const float*)d_in[13]; const float* ddqT = (const float*)d_in[14];
  const float* wfreq = (const float*)d_in[15];
  const float* wdamp = (const float*)d_in[16];
  const float* wphase = (const float*)d_in[17];
  const float* glW  = (const float*)d_in[18]; const float* glb = (const float*)d_in[19];
  const float* grW  = (const float*)d_in[20]; const float* grb = (const float*)d_in[21];
  const float* pq   = (const float*)d_in[22];
  const float* lnG  = (const float*)d_in[23]; const float* lnB = (const float*)d_in[24];
  const float* skA  = (const float*)d_in[25]; const float* skW = (const float*)d_in[26];
  const float* coup = (const float*)d_in[27];
  const int* cov_idx = (const int*)d_in[28];
  const int* cov_depth = (const int*)d_in[29];
  const unsigned char* cov_mask = (const unsigned char*)d_in[30];
  const int* kvalid = (const int*)d_in[31];
  const int S = in_sizes[28] / N_;

  // workspace carve-up (floats)
  float* ws = (float*)d_ws;
  const size_t P1 = (size_t)B_ * N_ * E_; // 1,048,576
  float* q_     = ws;
  float* v_     = q_ + P1;
  float* kl_    = v_ + P1;
  float* vl_    = kl_ + P1;
  float* gpre_  = vl_ + P1;
  float* local_ = gpre_ + P1;
  float* hidden_ = local_ + P1;
  float* Qall_  = hidden_ + P1;                               // DP * B*N*H*DH
  float* bank_  = Qall_ + (size_t)DP_ * B_ * N_ * H_ * DH_;   // B*H*NODES*KMAX*DH

  const dim3 gemm_grid(B_ * N_ / 16, E_ / 64);
  proj_gemm<<<gemm_grid, 128, 0, stream>>>(x, qW, qb, q_, E_, E_);
  proj_gemm<<<gemm_grid, 128, 0, stream>>>(x, vW, vb, v_, E_, E_);
  proj_gemm<<<gemm_grid, 128, 0, stream>>>(x, klW, klb, kl_, E_, E_);
  proj_gemm<<<gemm_grid, 128, 0, stream>>>(x, vlW, vlb, vl_, E_, E_);
  proj_gemm<<<gemm_grid, 128, 0, stream>>>(x, gW, gb, gpre_, E_, E_);

  scatter_leaves<<<(B_ * N_ * E_) / 256, 256, 0, stream>>>(v_, bank_);
  local_attn<<<dim3(N_, B_), 256, 0, stream>>>(q_, kl_, vl_, local_);

  int offs[12];
  offs[0] = 0;
  for (int d = 1; d <= 11; ++d) offs[d] = offs[d - 1] + (N_ >> (d - 1));

  level_kernel<1, 2><<<dim3(N_ >> 1, B_), 256, 0, stream>>>(
      bank_, 1, offs[0], offs[1], wfreq, wdamp, wphase, glW, glb, grW, grb, pq, lnG, lnB, skA, skW, coup);
  level_kernel<2, 4><<<dim3(N_ >> 2, B_), 256, 0, stream>>>(
      bank_, 2, offs[1], offs[2], wfreq, wdamp, wphase, glW, glb, grW, grb, pq, lnG, lnB, skA, skW, coup);
  level_kernel<4, 8><<<dim3(N_ >> 3, B_), 256, 0, stream>>>(
      bank_, 3, offs[2], offs[3], wfreq, wdamp, wphase, glW, glb, grW, grb, pq, lnG, lnB, skA, skW, coup);
  for (int d = 4; d <= 10; ++d) {
    level_kernel<8, 8><<<dim3(N_ >> d, B_), 256, 0, stream>>>(
        bank_, d, offs[d - 1], offs[d], wfreq, wdamp, wphase, glW, glb, grW, grb, pq, lnG, lnB, skA, skW, coup);
  }

  qall_gemm<<<dim3(B_ * N_ * H_ / 16, DP_), 128, 0, stream>>>(q_, ddqW, Qall_);

  tree_combine<<<dim3(N_, B_), 256, 0, stream>>>(Qall_, bank_, local_, gpre_, ddqT,
                                                 cov_idx, cov_depth, cov_mask, kvalid,
                                                 hidden_, S);

  proj_gemm<<<gemm_grid, 128, 0, stream>>>(hidden_, oW, ob, (float*)d_out, E_, E_);
}